// DistSAGE_13065290515269
// MI455X (gfx1250) — compile-verified
//
#include <hip/hip_runtime.h>
#include <hip/hip_bf16.h>

typedef __attribute__((ext_vector_type(2))) float v2f;
typedef __attribute__((ext_vector_type(8))) float v8f;

#define N_SRC0 300000
#define N_DST0 100000
#define N_DST1 20000
#define N_DST2 4096
#define E0 1000000
#define E1 300000
#define E2 61440
#define D_IN 128
#define D_H  256
#define D_OUT 47

// ---------------------------------------------------------------------------
// Zero-fill (accumulator regions of workspace)
// ---------------------------------------------------------------------------
__global__ void zero_f32(float* __restrict__ p, long long n) {
    long long i = (long long)blockIdx.x * blockDim.x + threadIdx.x;
    if (i < n) p[i] = 0.0f;
}

// ---------------------------------------------------------------------------
// Weight transpose: WT[n*K + k] = W[k*N + n]   (runs once per launch, tiny)
// ---------------------------------------------------------------------------
__global__ void transpose_w(const float* __restrict__ W, float* __restrict__ WT,
                            int K, int N) {
    int i = blockIdx.x * blockDim.x + threadIdx.x;
    if (i < K * N) {
        int k = i / N, n = i - k * N;
        WT[(size_t)n * K + k] = W[i];
    }
}

// ---------------------------------------------------------------------------
// Edge scatter-add: agg[dst[e], :] += feat[src[e], :], deg[dst[e]] += 1
// One thread handles 4 consecutive features (b128 load + 4 f32 atomics).
// ---------------------------------------------------------------------------
__global__ void scatter_add(const float* __restrict__ feat,
                            const int* __restrict__ src,
                            const int* __restrict__ dst,
                            float* __restrict__ agg,
                            float* __restrict__ deg,
                            long long total_chunks, int dShift) {
    long long tid = (long long)blockIdx.x * blockDim.x + threadIdx.x;
    if (tid >= total_chunks) return;
    const int chunksPerRow = 1 << (dShift - 2);
    const int e = (int)(tid >> (dShift - 2));
    const int f = (int)(tid & (chunksPerRow - 1)) << 2;
    const int s = src[e];
    const int d = dst[e];
    const float4 v = *(const float4*)(feat + ((size_t)s << dShift) + f);
    float* a = agg + ((size_t)d << dShift) + f;
    atomicAdd(a + 0, v.x);
    atomicAdd(a + 1, v.y);
    atomicAdd(a + 2, v.z);
    atomicAdd(a + 3, v.w);
    if (f == 0) atomicAdd(deg + d, 1.0f);
}

// ---------------------------------------------------------------------------
// Fused SAGE layer GEMM, 16x64 output strip per wave, compile-time K:
//   Out[m,n] = act( Aself[m,:]@Ws[:,n] + (Agg[m,:]/max(Deg[m],1))@Wn[:,n] + b[n] )
// Weights pre-transposed: WT[n*K + k] -> each B fragment is one b64 load at
// (base + immediate offset). With K a template constant and unroll-by-8, the
// inner loop has no per-load address math: 10 base pointers advance once per
// 8 k-steps, loads use growing immediate offsets, and the scheduler can batch
// them into clauses ahead of the 64 WMMAs in the window.
//
// Fragment layouts per CDNA5 ISA 7.12.2 (wave32):
//   A 16x4 : lane = 16*hi + m ; VGPR j holds A[m][k0 + 2*hi + j]
//   B 4x16 : lane = 16*hi + n ; VGPR j holds B[k0 + 2*hi + j][n]
//   C 16x16: lane = 16*hi + n ; VGPR v holds C[v + 8*hi][n]
// ---------------------------------------------------------------------------
template<int K>
__global__ __launch_bounds__(256)
void sage_gemm_wmma(const float* __restrict__ Aself,
                    const float* __restrict__ Agg,
                    const float* __restrict__ Deg,
                    const float* __restrict__ WsT,
                    const float* __restrict__ WnT,
                    const float* __restrict__ Bias,
                    float* __restrict__ Out,
                    int M, int Nout, int relu) {
    const int lane = threadIdx.x & 31;
    const int wave = threadIdx.x >> 5;
    const int tilesNG = (Nout + 63) >> 6;       // 64-wide N groups
    const int tilesM = M >> 4;
    const int tileIdx = blockIdx.x * (blockDim.x >> 5) + wave;
    if (tileIdx >= tilesM * tilesNG) return;    // wave-uniform: EXEC all-1s

    const int tm = tileIdx / tilesNG;
    const int tg = tileIdx - tm * tilesNG;
    const int ml = lane & 15;                   // m for A, n for B/C
    const int hi = lane >> 4;                   // lane-half select
    const int kb = hi << 1;                     // per-lane K phase

    const int rowA = tm * 16 + ml;
    const float invd = 1.0f / fmaxf(Deg[rowA], 1.0f);

    // All stream pointers pre-offset by the per-lane K phase: inner loop
    // indexes them with compile-time-constant k0 only.
    const float* aS = Aself + (size_t)rowA * K + kb;
    const float* aN = Agg   + (size_t)rowA * K + kb;

    int col[4];
    const float* pS[4];
    const float* pN[4];
    #pragma unroll
    for (int t = 0; t < 4; ++t) {
        col[t] = tg * 64 + t * 16 + ml;
        const int cc = col[t] < Nout ? col[t] : (Nout - 1);  // clamp for loads
        pS[t] = WsT + (size_t)cc * K + kb;
        pN[t] = WnT + (size_t)cc * K + kb;
    }

    v8f c0 = {0.f,0.f,0.f,0.f,0.f,0.f,0.f,0.f};
    v8f c1 = c0, c2 = c0, c3 = c0;

    #pragma unroll 8
    for (int k0 = 0; k0 < K; k0 += 4) {
        v2f as, an, b0, b1, b2, b3, n0, n1, n2, n3;
        as[0] = aS[k0];     as[1] = aS[k0 + 1];
        an[0] = aN[k0];     an[1] = aN[k0 + 1];
        b0[0] = pS[0][k0];  b0[1] = pS[0][k0 + 1];
        b1[0] = pS[1][k0];  b1[1] = pS[1][k0 + 1];
        b2[0] = pS[2][k0];  b2[1] = pS[2][k0 + 1];
        b3[0] = pS[3][k0];  b3[1] = pS[3][k0 + 1];
        n0[0] = pN[0][k0];  n0[1] = pN[0][k0 + 1];
        n1[0] = pN[1][k0];  n1[1] = pN[1][k0 + 1];
        n2[0] = pN[2][k0];  n2[1] = pN[2][k0 + 1];
        n3[0] = pN[3][k0];  n3[1] = pN[3][k0 + 1];

        v2f anX;
        anX[0] = an[0] * invd;
        anX[1] = an[1] * invd;

        c0 = __builtin_amdgcn_wmma_f32_16x16x4_f32(false, as,  false, b0, (short)0, c0, false, false);
        c1 = __builtin_amdgcn_wmma_f32_16x16x4_f32(false, as,  false, b1, (short)0, c1, false, false);
        c2 = __builtin_amdgcn_wmma_f32_16x16x4_f32(false, as,  false, b2, (short)0, c2, false, false);
        c3 = __builtin_amdgcn_wmma_f32_16x16x4_f32(false, as,  false, b3, (short)0, c3, false, false);
        c0 = __builtin_amdgcn_wmma_f32_16x16x4_f32(false, anX, false, n0, (short)0, c0, false, false);
        c1 = __builtin_amdgcn_wmma_f32_16x16x4_f32(false, anX, false, n1, (short)0, c1, false, false);
        c2 = __builtin_amdgcn_wmma_f32_16x16x4_f32(false, anX, false, n2, (short)0, c2, false, false);
        c3 = __builtin_amdgcn_wmma_f32_16x16x4_f32(false, anX, false, n3, (short)0, c3, false, false);
    }

    v8f acc[4] = {c0, c1, c2, c3};
    #pragma unroll
    for (int t = 0; t < 4; ++t) {
        if (col[t] < Nout) {
            const float bias = Bias[col[t]];
            #pragma unroll
            for (int v = 0; v < 8; ++v) {
                float val = acc[t][v] + bias;
                if (relu) val = fmaxf(val, 0.0f);
                const int row = tm * 16 + v + (hi << 3);
                Out[(size_t)row * Nout + col[t]] = val;
            }
        }
    }
}

// ---------------------------------------------------------------------------
// Host-side orchestration
// ---------------------------------------------------------------------------
extern "C" void kernel_launch(void* const* d_in, const int* in_sizes, int n_in,
                              void* d_out, int out_size, void* d_ws, size_t ws_size,
                              hipStream_t stream) {
    const float* x    = (const float*)d_in[0];
    const int*   src0 = (const int*)  d_in[1];
    const int*   dst0 = (const int*)  d_in[2];
    const int*   src1 = (const int*)  d_in[3];
    const int*   dst1 = (const int*)  d_in[4];
    const int*   src2 = (const int*)  d_in[5];
    const int*   dst2 = (const int*)  d_in[6];
    const float* Ws0  = (const float*)d_in[7];
    const float* Wn0  = (const float*)d_in[8];
    const float* b0   = (const float*)d_in[9];
    const float* Ws1  = (const float*)d_in[10];
    const float* Wn1  = (const float*)d_in[11];
    const float* b1   = (const float*)d_in[12];
    const float* Ws2  = (const float*)d_in[13];
    const float* Wn2  = (const float*)d_in[14];
    const float* b2   = (const float*)d_in[15];
    float* out = (float*)d_out;

    // Workspace layout (floats):
    //   [agg0|deg0|agg1|deg1|agg2|deg2 | WT x6 | h1 | h0]
    float* ws    = (float*)d_ws;
    float* agg0  = ws;                                  // 100000*128
    float* deg0  = agg0 + (size_t)N_DST0 * D_IN;        // 100000
    float* agg1  = deg0 + N_DST0;                       // 20000*256
    float* deg1  = agg1 + (size_t)N_DST1 * D_H;         // 20000
    float* agg2  = deg1 + N_DST1;                       // 4096*256
    float* deg2  = agg2 + (size_t)N_DST2 * D_H;         // 4096
    float* Ws0T  = deg2 + N_DST2;                       // 256*128
    float* Wn0T  = Ws0T + D_IN * D_H;
    float* Ws1T  = Wn0T + D_IN * D_H;                   // 256*256
    float* Wn1T  = Ws1T + D_H * D_H;
    float* Ws2T  = Wn1T + D_H * D_H;                    // 47*256
    float* Wn2T  = Ws2T + D_H * D_OUT;
    float* h1    = Wn2T + D_H * D_OUT;                  // 20000*256
    float* h0    = h1 + (size_t)N_DST1 * D_H;           // 100000*256

    const long long nzero = (long long)N_DST0 * D_IN + N_DST0
                          + (long long)N_DST1 * D_H + N_DST1
                          + (long long)N_DST2 * D_H + N_DST2;
    zero_f32<<<(unsigned)((nzero + 255) / 256), 256, 0, stream>>>(ws, nzero);

    // Transpose all six weight matrices up front.
    transpose_w<<<(D_IN * D_H + 255) / 256, 256, 0, stream>>>(Ws0, Ws0T, D_IN, D_H);
    transpose_w<<<(D_IN * D_H + 255) / 256, 256, 0, stream>>>(Wn0, Wn0T, D_IN, D_H);
    transpose_w<<<(D_H * D_H + 255) / 256, 256, 0, stream>>>(Ws1, Ws1T, D_H, D_H);
    transpose_w<<<(D_H * D_H + 255) / 256, 256, 0, stream>>>(Wn1, Wn1T, D_H, D_H);
    transpose_w<<<(D_H * D_OUT + 255) / 256, 256, 0, stream>>>(Ws2, Ws2T, D_H, D_OUT);
    transpose_w<<<(D_H * D_OUT + 255) / 256, 256, 0, stream>>>(Wn2, Wn2T, D_H, D_OUT);

    // ---- Layer 0 ----
    {
        long long chunks = (long long)E0 * (D_IN / 4);
        scatter_add<<<(unsigned)((chunks + 255) / 256), 256, 0, stream>>>(
            x, src0, dst0, agg0, deg0, chunks, 7 /*log2 128*/);
        int tiles = (N_DST0 / 16) * ((D_H + 63) / 64);   // 6250 * 4
        sage_gemm_wmma<D_IN><<<(unsigned)((tiles + 7) / 8), 256, 0, stream>>>(
            x, agg0, deg0, Ws0T, Wn0T, b0, h0, N_DST0, D_H, 1);
    }
    // ---- Layer 1 ----
    {
        long long chunks = (long long)E1 * (D_H / 4);
        scatter_add<<<(unsigned)((chunks + 255) / 256), 256, 0, stream>>>(
            h0, src1, dst1, agg1, deg1, chunks, 8 /*log2 256*/);
        int tiles = (N_DST1 / 16) * ((D_H + 63) / 64);   // 1250 * 4
        sage_gemm_wmma<D_H><<<(unsigned)((tiles + 7) / 8), 256, 0, stream>>>(
            h0, agg1, deg1, Ws1T, Wn1T, b1, h1, N_DST1, D_H, 1);
    }
    // ---- Layer 2 ----
    {
        long long chunks = (long long)E2 * (D_H / 4);
        scatter_add<<<(unsigned)((chunks + 255) / 256), 256, 0, stream>>>(
            h1, src2, dst2, agg2, deg2, chunks, 8 /*log2 256*/);
        int tiles = (N_DST2 / 16) * ((D_OUT + 63) / 64); // 256 * 1
        sage_gemm_wmma<D_H><<<(unsigned)((tiles + 7) / 8), 256, 0, stream>>>(
            h1, agg2, deg2, Ws2T, Wn2T, b2, out, N_DST2, D_OUT, 0);
    }
}